// ClusterMemory_teacher_37366215475659
// MI455X (gfx1250) — compile-verified
//
#include <hip/hip_runtime.h>
#include <hip/hip_bf16.h>
#include <math.h>

typedef __attribute__((ext_vector_type(2))) float v2f;
typedef __attribute__((ext_vector_type(8))) float v8f;

#define B_ROWS 256
#define N_FEAT 8192
#define D_DIM  2048
#define INV_T  20.0f   // 1/0.05

// ---------------------------------------------------------------------------
// Kernel 1: normalize x rows; xn2[b] = ||xn_row||^2 (faithful to reference)
// ---------------------------------------------------------------------------
__global__ __launch_bounds__(256) void norm_rows(const float* __restrict__ x,
                                                 float* __restrict__ xn,
                                                 float* __restrict__ xn2) {
    __shared__ float red[256];
    const int b = blockIdx.x;
    const int tid = threadIdx.x;
    const float* xr = x + (size_t)b * D_DIM;
    float s = 0.0f;
    for (int j = tid; j < D_DIM; j += 256) { float v = xr[j]; s += v * v; }
    red[tid] = s; __syncthreads();
    for (int off = 128; off > 0; off >>= 1) {
        if (tid < off) red[tid] += red[tid + off];
        __syncthreads();
    }
    const float ss  = red[0];
    const float inv = 1.0f / fmaxf(sqrtf(ss), 1e-12f);
    for (int j = tid; j < D_DIM; j += 256) xn[(size_t)b * D_DIM + j] = xr[j] * inv;
    if (tid == 0) xn2[b] = ss * inv * inv;
}

// ---------------------------------------------------------------------------
// Kernel 2: f2[n] = sum(f[n,:]^2)
// ---------------------------------------------------------------------------
__global__ __launch_bounds__(256) void rowsq(const float* __restrict__ f,
                                             float* __restrict__ f2) {
    __shared__ float red[256];
    const int n = blockIdx.x;
    const int tid = threadIdx.x;
    const float* fr = f + (size_t)n * D_DIM;
    float s = 0.0f;
    for (int j = tid; j < D_DIM; j += 256) { float v = fr[j]; s += v * v; }
    red[tid] = s; __syncthreads();
    for (int off = 128; off > 0; off >>= 1) {
        if (tid < off) red[tid] += red[tid + off];
        __syncthreads();
    }
    if (tid == 0) f2[n] = red[0];
}

// ---------------------------------------------------------------------------
// Kernel 3: G = xn @ f^T via V_WMMA_F32_16X16X4_F32 (full fp32 matrix path).
// One wave: M tile (16 rows of x) x 4 N tiles (64 rows of f); K loop over 2048.
// A/B fragment layout (ISA 7.12.2, 32-bit A 16x4): lane%16 = row,
// (lane/16)*2 + v = K within the 4-wide K slab -> one b64 load per fragment.
// ---------------------------------------------------------------------------
__global__ __launch_bounds__(256) void gemm_wmma(const float* __restrict__ xn,
                                                 const float* __restrict__ f,
                                                 float* __restrict__ G) {
    const int lane = threadIdx.x & 31;
    const int wave = threadIdx.x >> 5;
    const int w    = blockIdx.x * 8 + wave;   // 0..2047
    const int mt   = w & 15;                  // 16 M tiles of 16
    const int ng   = w >> 4;                  // 128 N groups of 64

    const int row  = lane & 15;
    const int ksel = (lane >> 4) << 1;        // 0 or 2

    const float* ap = xn + (size_t)(mt * 16 + row) * D_DIM + ksel;
    const float* bp = f  + (size_t)(ng * 64 + row) * D_DIM + ksel;

    v8f c0 = {}, c1 = {}, c2 = {}, c3 = {};

    #pragma unroll 4
    for (int k = 0; k < D_DIM; k += 4) {
        v2f a  = *(const v2f*)(ap + k);
        v2f b0 = *(const v2f*)(bp + k);
        v2f b1 = *(const v2f*)(bp + 16 * D_DIM + k);
        v2f b2 = *(const v2f*)(bp + 32 * D_DIM + k);
        v2f b3 = *(const v2f*)(bp + 48 * D_DIM + k);
        c0 = __builtin_amdgcn_wmma_f32_16x16x4_f32(false, a, false, b0, (short)0, c0, false, false);
        c1 = __builtin_amdgcn_wmma_f32_16x16x4_f32(false, a, false, b1, (short)0, c1, false, false);
        c2 = __builtin_amdgcn_wmma_f32_16x16x4_f32(false, a, false, b2, (short)0, c2, false, false);
        c3 = __builtin_amdgcn_wmma_f32_16x16x4_f32(false, a, false, b3, (short)0, c3, false, false);
    }

    // C/D layout: VGPR v, lanes 0-15 -> M=v, lanes 16-31 -> M=v+8; N = lane&15
    const int mbase = mt * 16 + ((lane >> 4) << 3);
    const int ncol  = lane & 15;
    float* g0 = G + (size_t)mbase * N_FEAT + ng * 64 + ncol;
    #pragma unroll
    for (int v = 0; v < 8; ++v) {
        g0[(size_t)v * N_FEAT +  0] = c0[v];
        g0[(size_t)v * N_FEAT + 16] = c1[v];
        g0[(size_t)v * N_FEAT + 32] = c2[v];
        g0[(size_t)v * N_FEAT + 48] = c3[v];
    }
}

// ---------------------------------------------------------------------------
// Block reduction helpers (deterministic fixed-order trees)
// ---------------------------------------------------------------------------
__device__ __forceinline__ float block_max(float v, float* red) {
    const int tid = threadIdx.x;
    red[tid] = v; __syncthreads();
    for (int off = 128; off > 0; off >>= 1) {
        if (tid < off) red[tid] = fmaxf(red[tid], red[tid + off]);
        __syncthreads();
    }
    float r = red[0]; __syncthreads();
    return r;
}
__device__ __forceinline__ float block_sum(float v, float* red) {
    const int tid = threadIdx.x;
    red[tid] = v; __syncthreads();
    for (int off = 128; off > 0; off >>= 1) {
        if (tid < off) red[tid] += red[tid + off];
        __syncthreads();
    }
    float r = red[0]; __syncthreads();
    return r;
}

// ---------------------------------------------------------------------------
// Kernel 4: per-row losses. One block per row b; G row staged in LDS (32KB).
//  CE1 = lse(G/T) - G[t]/T
//  dist = sqrt(max(xn2 + f2 - 2G, 0)); soft = softmax(dist)
//  CE2 = log(sum_j exp(soft_j)) - soft_t
// ---------------------------------------------------------------------------
__global__ __launch_bounds__(256) void row_loss(const float* __restrict__ G,
                                                const float* __restrict__ f2,
                                                const float* __restrict__ xn2,
                                                const int* __restrict__ targets,
                                                float* __restrict__ rl,
                                                int branch) {
    __shared__ float sG[N_FEAT];
    __shared__ float red[256];
    const int b = blockIdx.x;
    const int tid = threadIdx.x;
    const float* gr = G + (size_t)b * N_FEAT;
    for (int j = tid; j < N_FEAT; j += 256) sG[j] = gr[j];
    __syncthreads();
    const float x2 = xn2[b];

    // pass 1: maxima for stable reductions
    float m1 = -INFINITY, md = -INFINITY;
    for (int j = tid; j < N_FEAT; j += 256) {
        float g = sG[j];
        m1 = fmaxf(m1, g * INV_T);
        float d = sqrtf(fmaxf(x2 + f2[j] - 2.0f * g, 0.0f));
        md = fmaxf(md, d);
    }
    m1 = block_max(m1, red);
    md = block_max(md, red);

    // pass 2: partition sums
    float z1 = 0.0f, zd = 0.0f;
    for (int j = tid; j < N_FEAT; j += 256) {
        float g = sG[j];
        z1 += expf(g * INV_T - m1);
        float d = sqrtf(fmaxf(x2 + f2[j] - 2.0f * g, 0.0f));
        zd += expf(d - md);
    }
    z1 = block_sum(z1, red);
    zd = block_sum(zd, red);
    const float inv_zd = 1.0f / zd;

    // pass 3: S2 = sum exp(softmax(dist))
    float s2 = 0.0f;
    for (int j = tid; j < N_FEAT; j += 256) {
        float g = sG[j];
        float d = sqrtf(fmaxf(x2 + f2[j] - 2.0f * g, 0.0f));
        s2 += expf(expf(d - md) * inv_zd);
    }
    s2 = block_sum(s2, red);

    if (tid == 0) {
        const int t = targets[b];
        const float gt = sG[t];
        const float dt = sqrtf(fmaxf(x2 + f2[t] - 2.0f * gt, 0.0f));
        const float ce1 = m1 + logf(z1) - gt * INV_T;
        const float soft_t = expf(dt - md) * inv_zd;
        const float ce2 = logf(s2) - soft_t;
        rl[branch * B_ROWS + b] = ce1 + ce2;
    }
}

// ---------------------------------------------------------------------------
// Kernel 5: deterministic final sum: loss = 0.5/256 * sum(rl[0..767])
// ---------------------------------------------------------------------------
__global__ __launch_bounds__(256) void final_sum(const float* __restrict__ rl,
                                                 float* __restrict__ out) {
    __shared__ float red[256];
    const int tid = threadIdx.x;
    float s = rl[tid] + rl[tid + 256] + rl[tid + 512];
    red[tid] = s; __syncthreads();
    for (int off = 128; off > 0; off >>= 1) {
        if (tid < off) red[tid] += red[tid + off];
        __syncthreads();
    }
    if (tid == 0) out[0] = red[0] * (0.5f / 256.0f);
}

// ---------------------------------------------------------------------------
extern "C" void kernel_launch(void* const* d_in, const int* in_sizes, int n_in,
                              void* d_out, int out_size, void* d_ws, size_t ws_size,
                              hipStream_t stream) {
    const float* xin[3] = { (const float*)d_in[0], (const float*)d_in[1], (const float*)d_in[2] };
    const int* targets  = (const int*)d_in[3];
    // d_in[4] = epoch (unused)
    const float* fin[3] = { (const float*)d_in[5], (const float*)d_in[6], (const float*)d_in[7] };

    char* ws = (char*)d_ws;
    float* G   = (float*)(ws);                                          // 8 MB
    float* xn  = (float*)(ws + (size_t)B_ROWS * N_FEAT * sizeof(float));// 2 MB
    float* f2  = (float*)((char*)xn + (size_t)B_ROWS * D_DIM * sizeof(float));
    float* xn2 = (float*)((char*)f2 + (size_t)N_FEAT * sizeof(float));
    float* rl  = (float*)((char*)xn2 + (size_t)B_ROWS * sizeof(float));

    for (int br = 0; br < 3; ++br) {
        norm_rows<<<B_ROWS, 256, 0, stream>>>(xin[br], xn, xn2);
        rowsq<<<N_FEAT, 256, 0, stream>>>(fin[br], f2);
        gemm_wmma<<<256, 256, 0, stream>>>(xn, fin[br], G);
        row_loss<<<B_ROWS, 256, 0, stream>>>(G, f2, xn2, targets, rl, br);
    }
    final_sum<<<1, 256, 0, stream>>>(rl, (float*)d_out);
}